// GCN_Layer_30262339568119
// MI455X (gfx1250) — compile-verified
//
#include <hip/hip_runtime.h>

// ---------------------------------------------------------------------------
// GCN layer for MI455X (gfx1250, wave32):
//   agg = scatter-add over edges (SpMM)            -> kernel 1 (f32 atomics)
//   out = leakyrelu((agg+x)@W1^T + b1 + (agg*x)@W2^T + b2)
//                                                  -> kernel 2 (WMMA f32 16x16x4)
// ---------------------------------------------------------------------------

typedef float v2f __attribute__((ext_vector_type(2)));
typedef float v8f __attribute__((ext_vector_type(8)));

#define FDIM 64
#define NEG_SLOPE 0.01f

// ------------------------- SpMM scatter kernel -----------------------------
// 16 lanes per edge; each lane: one b128 gather + 4 native f32 atomics.
__global__ __launch_bounds__(256) void gcn_spmm_kernel(
    const float* __restrict__ feat,
    const int*   __restrict__ erow,
    const int*   __restrict__ ecol,
    const float* __restrict__ eval,
    float*       __restrict__ agg,
    int nEdges)
{
    long long tid = (long long)blockIdx.x * 256 + threadIdx.x;
    int e = (int)(tid >> 4);
    if (e >= nEdges) return;
    int l = (int)(tid & 15);

    int   r = erow[e];
    int   c = ecol[e];
    float v = eval[e];

    const float4 f = *((const float4*)(feat + (size_t)c * FDIM) + l);
    float* dst = agg + (size_t)r * FDIM + l * 4;

    unsafeAtomicAdd(dst + 0, v * f.x);
    unsafeAtomicAdd(dst + 1, v * f.y);
    unsafeAtomicAdd(dst + 2, v * f.z);
    unsafeAtomicAdd(dst + 3, v * f.w);
}

// --------------------- Fused dual-GEMM + epilogue --------------------------
// One wave per 16-row tile. A1 = agg + x, A2 = agg .* x (16x64 each).
// out_tile(16x64) = A1 @ W1^T + A2 @ W2^T + (b1+b2), LeakyReLU.
// WMMA f32 16x16x4 layouts (ISA 7.12.2):
//   A frag (v2f): row M = lane%16, cols k0 + 2*(lane/16) + {0,1}
//   B frag (v2f): B[k][n] = W[n][k] -> same indexing on W
//   C/D   (v8f): element v -> M = v + 8*(lane/16), N = lane%16
__global__ __launch_bounds__(256) void gcn_gemm_kernel(
    const float* __restrict__ agg,
    const float* __restrict__ x,
    const float* __restrict__ W1,
    const float* __restrict__ b1,
    const float* __restrict__ W2,
    const float* __restrict__ b2,
    float*       __restrict__ out,
    int nRows)
{
    const int lane = threadIdx.x & 31;
    const int wave = threadIdx.x >> 5;
    const int tile = blockIdx.x * 8 + wave;
    if (tile * 16 >= nRows) return;          // wave-uniform: EXEC stays all-1s

    const int m    = lane & 15;              // row / B-column within tile
    const int koff = (lane >> 4) * 2;        // 0 or 2

    // --- Preload A fragments for all 16 K-chunks (A1 = agg+x, A2 = agg*x) ---
    const float* aggRow = agg + (size_t)(tile * 16 + m) * FDIM;
    const float* xRow   = x   + (size_t)(tile * 16 + m) * FDIM;

    v2f a1[16], a2[16];
#pragma unroll
    for (int kc = 0; kc < 16; ++kc) {
        int k0 = kc * 4 + koff;              // even -> 8B aligned
        v2f g = *(const v2f*)(aggRow + k0);
        v2f f = *(const v2f*)(xRow   + k0);
        a1[kc] = g + f;
        a2[kc] = g * f;
    }

    v8f acc[4] = {};                         // 4 column tiles of 16

#pragma unroll
    for (int kc = 0; kc < 16; ++kc) {
        int k0 = kc * 4 + koff;
#pragma unroll
        for (int nt = 0; nt < 4; ++nt) {
            const size_t wrow = (size_t)(nt * 16 + m) * FDIM + k0;
            v2f bw1 = *(const v2f*)(W1 + wrow);
            v2f bw2 = *(const v2f*)(W2 + wrow);
            acc[nt] = __builtin_amdgcn_wmma_f32_16x16x4_f32(
                false, a1[kc], false, bw1, (short)0, acc[nt], false, false);
            acc[nt] = __builtin_amdgcn_wmma_f32_16x16x4_f32(
                false, a2[kc], false, bw2, (short)0, acc[nt], false, false);
        }
    }

    // ------------------------------ Epilogue -------------------------------
    const int mhalf = (lane >> 4) * 8;       // +0 or +8 row offset for C/D
#pragma unroll
    for (int nt = 0; nt < 4; ++nt) {
        const int col  = nt * 16 + m;        // N = lane%16
        const float bs = b1[col] + b2[col];
#pragma unroll
        for (int v = 0; v < 8; ++v) {
            const int outRow = tile * 16 + v + mhalf;
            float val = acc[nt][v] + bs;
            val = (val >= 0.0f) ? val : NEG_SLOPE * val;
            out[(size_t)outRow * FDIM + col] = val;
        }
    }
}

// ------------------------------- Launcher ----------------------------------
extern "C" void kernel_launch(void* const* d_in, const int* in_sizes, int n_in,
                              void* d_out, int out_size, void* d_ws, size_t ws_size,
                              hipStream_t stream)
{
    const float* feat = (const float*)d_in[0];
    const int*   erow = (const int*)  d_in[1];
    const int*   ecol = (const int*)  d_in[2];
    const float* eval = (const float*)d_in[3];
    const float* W1   = (const float*)d_in[4];
    const float* b1   = (const float*)d_in[5];
    const float* W2   = (const float*)d_in[6];
    const float* b2   = (const float*)d_in[7];

    const int nNodes = in_sizes[0] / FDIM;   // 100000
    const int nEdges = in_sizes[1];          // 1600000

    float* agg = (float*)d_ws;               // N*64 f32 = 25.6 MB scratch
    hipMemsetAsync(agg, 0, (size_t)nNodes * FDIM * sizeof(float), stream);

    // SpMM: 16 lanes per edge
    long long spmmThreads = (long long)nEdges * 16;
    int spmmBlocks = (int)((spmmThreads + 255) / 256);
    gcn_spmm_kernel<<<spmmBlocks, 256, 0, stream>>>(feat, erow, ecol, eval,
                                                    agg, nEdges);

    // Fused dual-GEMM epilogue: one wave per 16-row tile, 8 waves per block
    int tiles      = (nNodes + 15) / 16;     // 6250 (exact)
    int gemmBlocks = (tiles + 7) / 8;
    gcn_gemm_kernel<<<gemmBlocks, 256, 0, stream>>>(agg, feat, W1, b1, W2, b2,
                                                    (float*)d_out, nNodes);
}